// RandomProjectionQuantizer_59115929862696
// MI455X (gfx1250) — compile-verified
//
#include <hip/hip_runtime.h>
#include <hip/hip_bf16.h>

// ---------------------------------------------------------------------------
// RandomProjectionQuantizer for MI455X (gfx1250), wave32 — single fused kernel.
//
// targets = X(32768x320) @ P(320x16); row-normalize; labels = argmax over
// 8192 unit codebook vectors of targets_norm @ CB^T (== argmin euclidean).
//
// Roofline: X read = 41.9 MB @ 23.3 TB/s ~= 1.8 us (HBM term). Codebook is
// streamed from L2 once PER BLOCK (512 KB): with 32 rows/wave x 8 waves =
// 256 rows/block -> 128 blocks -> 64 MB of L2 traffic, keeping the L2 stream
// below the HBM term. Compute is only ~9 GFLOP, so fp32 WMMA
// (V_WMMA_F32_16X16X4_F32) is the right precision — matrix throughput is free.
// All staging is gfx1250 GLOBAL_LOAD_ASYNC_TO_LDS_B128 (ASYNCcnt), double
// buffered so copies overlap WMMAs; one barrier per codebook stage.
//
// LDS/block: 64 KB X stage + 16 KB targets + 8 KB codebook = 88 KB (of 320 KB).
// ---------------------------------------------------------------------------

typedef __attribute__((ext_vector_type(2))) float v2f;
typedef __attribute__((ext_vector_type(8))) float v8f;

#define NROWS   32768   // B*L = 16*2048
#define DDIM    320     // stacked feature dim
#define KDIM    16      // codebook dim
#define NCODES  8192    // codebook size

// Async global->LDS copy, 16 B per lane, 64-bit VGPR address form.
__device__ __forceinline__ void async_copy_b128(unsigned lds_addr, const void* gptr) {
    unsigned long long ga = (unsigned long long)(uintptr_t)gptr;
    asm volatile("global_load_async_to_lds_b128 %0, %1, off"
                 :: "v"(lds_addr), "v"(ga) : "memory");
}

// Async global->LDS copy, SGPR base + 32-bit voffset form (GVS addressing).
__device__ __forceinline__ void async_copy_b128_s(unsigned lds_addr, unsigned voff,
                                                  const void* sbase) {
    asm volatile("global_load_async_to_lds_b128 %0, %1, %2"
                 :: "v"(lds_addr), "v"(voff), "s"(sbase) : "memory");
}

__global__ __launch_bounds__(256) void rpq_fused(
    const float* __restrict__ X,     // (NROWS, DDIM)
    const float* __restrict__ P,     // (DDIM, KDIM)
    const float* __restrict__ CB,    // (NCODES, KDIM) normalized
    int* __restrict__ labels)        // (NROWS)
{
    __shared__ __align__(16) float xs[8][2][16 * 64];    // 64 KB: X double buffer
    __shared__ __align__(16) float ts[8][2][16 * KDIM];  // 16 KB: normalized targets
    __shared__ __align__(16) float cs[2][64 * KDIM];     //  8 KB: codebook stages

    const int lane = threadIdx.x & 31;
    const int wid  = threadIdx.x >> 5;
    const int h    = lane >> 4;        // half-wave select
    const int n    = lane & 15;        // N / M-in-tile index

    // 32 rows per wave (two 16-row tile sets), 256 rows per block.
    const int row_base = blockIdx.x * 256 + wid * 32;

    // Single per-lane base for all P loads -> pure immediate offsets.
    const float* Pb = P + 2 * h * KDIM + n;

    // A fragments for the distance matmul, both row sets.
    v2f a[2][4];

    // ---------------- Phase 1: projection (2 x 16 rows per wave) ------------
    // A-frag (16x4 f32): lanes 0-15 = M=0..15; VGPR0 K=k0+2h, VGPR1 K=k0+2h+1.
    // B-frag (4x16 f32): mirrored K striping, N = lane&15.
    // C/D (16x16 f32): VGPR v -> row M = v + 8h, col N = lane&15.
    #pragma unroll
    for (int u = 0; u < 2; ++u) {
        const float* gX = X + (size_t)(row_base + u * 16) * DDIM;

        // Issue one 16x64 stage (1024 floats) as 8 async b128 ops per lane.
        auto issue_stage = [&](int s, int bsel) {
            #pragma unroll
            for (int i = 0; i < 8; ++i) {
                const int idx = i * 32 + lane;      // 0..255 float4 slots
                const int r   = idx >> 4;           // 0..15
                const int c4  = (idx & 15) * 4;     // 0..60
                const unsigned la =
                    (unsigned)(uintptr_t)&xs[wid][bsel][r * 64 + c4];
                async_copy_b128(la, gX + (size_t)r * DDIM + s * 64 + c4);
            }
        };

        issue_stage(0, 0);

        v8f acc = {};
        for (int s = 0; s < 5; ++s) {               // 5 stages of 64 columns
            if (s < 4) {
                issue_stage(s + 1, (s + 1) & 1);
                // 8 newer ops may stay outstanding; in-order retire => stage s done.
                asm volatile("s_wait_asynccnt 0x8" ::: "memory");
            } else {
                asm volatile("s_wait_asynccnt 0x0" ::: "memory");
            }
            const float* myxs = &xs[wid][s & 1][0];
            const float* Ps   = Pb + s * 64 * KDIM;

            #pragma unroll
            for (int kc = 0; kc < 16; ++kc) {       // 16 K-chunks of 4
                const int k0 = kc * 4 + 2 * h;
                v2f av, bv;
                av[0] = myxs[n * 64 + k0];
                av[1] = myxs[n * 64 + k0 + 1];
                bv[0] = Ps[(kc * 4) * KDIM];        // imm offset kc*256 B
                bv[1] = Ps[(kc * 4 + 1) * KDIM];    // imm offset kc*256+64 B
                acc = __builtin_amdgcn_wmma_f32_16x16x4_f32(
                    false, av, false, bv, (short)0, acc, false, false);
            }
        }

        // Normalize rows (row for VGPR v spans the 16-lane group) and park the
        // tile in LDS row-major: C/D layout -> A layout transpose via LDS.
        #pragma unroll
        for (int v = 0; v < 8; ++v) {
            float s2 = acc[v] * acc[v];
            #pragma unroll
            for (int off = 8; off > 0; off >>= 1)
                s2 += __shfl_xor(s2, off, 32);      // stays within 16-lane group
            const float scale = 1.0f / fmaxf(sqrtf(s2), 1e-12f);
            ts[wid][u][(v + 8 * h) * KDIM + n] = acc[v] * scale;
        }

        // A fragments (wave-private LDS, DS in-order guarantees RAW).
        #pragma unroll
        for (int kc = 0; kc < 4; ++kc) {
            const int k0 = kc * 4 + 2 * h;
            a[u][kc][0] = ts[wid][u][n * KDIM + k0];
            a[u][kc][1] = ts[wid][u][n * KDIM + k0 + 1];
        }
    }

    // ---------------- Phase 2: streaming argmax over the codebook ----------
    float best[2][8];
    int   bidx[2][8];
    #pragma unroll
    for (int u = 0; u < 2; ++u)
        #pragma unroll
        for (int v = 0; v < 8; ++v) { best[u][v] = -INFINITY; bidx[u][v] = 0; }

    // 32-bit voffset into CB (16 B per thread per stage), +0x1000 per stage.
    unsigned cb_voff = (unsigned)(threadIdx.x * 16);
    // LDS destinations toggle between the two 4 KB buffers.
    const unsigned cs_la0 = (unsigned)(uintptr_t)&cs[0][threadIdx.x * 4];
    const unsigned cs_la1 = (unsigned)(uintptr_t)&cs[1][threadIdx.x * 4];

    // Prologue: stage 0 -> buffer 0 (coalesced 4 KB per block).
    async_copy_b128_s(cs_la0, cb_voff, CB);
    cb_voff += 64 * KDIM * 4;

    for (int stage = 0; stage < 128; ++stage) {     // 128 stages of 64 codes
        const int b = stage & 1;
        asm volatile("s_wait_asynccnt 0x0" ::: "memory");  // own tile landed
        __syncthreads();  // all waves' tiles visible; old buffer fully consumed
        if (stage + 1 < 128) {
            async_copy_b128_s(b ? cs_la0 : cs_la1, cb_voff, CB);
            cb_voff += 64 * KDIM * 4;
        }

        #pragma unroll
        for (int t = 0; t < 4; ++t) {               // 4 N-tiles of 16 codes
            // B fragments loaded once, reused by both row sets.
            v2f bb[4];
            #pragma unroll
            for (int kc = 0; kc < 4; ++kc) {
                const int k0 = kc * 4 + 2 * h;
                bb[kc][0] = cs[b][(t * 16 + n) * KDIM + k0];
                bb[kc][1] = cs[b][(t * 16 + n) * KDIM + k0 + 1];
            }
            const int col = stage * 64 + t * 16 + n;

            #pragma unroll
            for (int u = 0; u < 2; ++u) {
                v8f c = {};
                #pragma unroll
                for (int kc = 0; kc < 4; ++kc)
                    c = __builtin_amdgcn_wmma_f32_16x16x4_f32(
                        false, a[u][kc], false, bb[kc], (short)0, c, false, false);
                #pragma unroll
                for (int v = 0; v < 8; ++v) {
                    // strict > : first max wins (matches argmin tie rule)
                    if (c[v] > best[u][v]) { best[u][v] = c[v]; bidx[u][v] = col; }
                }
            }
        }
    }

    // Cross-lane argmax within each 16-lane group (tie -> smaller index).
    #pragma unroll
    for (int u = 0; u < 2; ++u)
        #pragma unroll
        for (int v = 0; v < 8; ++v) {
            #pragma unroll
            for (int off = 8; off > 0; off >>= 1) {
                const float ob = __shfl_xor(best[u][v], off, 32);
                const int   oi = __shfl_xor(bidx[u][v], off, 32);
                if (ob > best[u][v] || (ob == best[u][v] && oi < bidx[u][v])) {
                    best[u][v] = ob; bidx[u][v] = oi;
                }
            }
            if (n == 0)
                labels[row_base + u * 16 + v + 8 * h] = bidx[u][v];
        }
}

// ---------------------------------------------------------------------------
extern "C" void kernel_launch(void* const* d_in, const int* in_sizes, int n_in,
                              void* d_out, int out_size, void* d_ws, size_t ws_size,
                              hipStream_t stream) {
    const float* X  = (const float*)d_in[0];   // (16,2048,320)
    const float* P  = (const float*)d_in[1];   // (320,16)
    const float* CB = (const float*)d_in[2];   // (8192,16)
    int* labels = (int*)d_out;                 // (16,2048) int32

    rpq_fused<<<NROWS / 256, 256, 0, stream>>>(X, P, CB, labels);
}